// GNNLayer_24550033064401
// MI455X (gfx1250) — compile-verified
//
#include <hip/hip_runtime.h>

typedef __attribute__((ext_vector_type(2))) float v2f;
typedef __attribute__((ext_vector_type(8))) float v8f;

#define IN_F 64

// Kernel 1: per 16-row tile, compute with f32 WMMA:
//   Q = X@W1          -> d_out = Q + (b1+b2)   (accumulator init for SpMM)
//   P = X@W1 + X²@W2  -> workspace             (gathered by SpMM pass)
// Block = 128 threads = 4 waves; wave w owns output columns [16w, 16w+16).
__global__ __launch_bounds__(128) void gnn_dense_wmma(
    const float* __restrict__ X, const float* __restrict__ W1,
    const float* __restrict__ b1, const float* __restrict__ W2,
    const float* __restrict__ b2, float* __restrict__ out,
    float* __restrict__ P, int nrows)
{
    __shared__ float ldsX[16 * IN_F];
    __shared__ float ldsX2[16 * IN_F];
    const int tid = threadIdx.x;
    const int rb  = blockIdx.x * 16;

    // Stage 16x64 X tile and its elementwise square into LDS (coalesced b128).
    for (int i = tid; i < (16 * IN_F) / 4; i += 128) {
        const int row  = i >> 4;          // (i*4)/64
        int srow = rb + row;
        if (srow > nrows - 1) srow = nrows - 1;   // clamp keeps EXEC full
        const int colq = (i & 15) * 4;
        const float4 xv = *(const float4*)(X + (size_t)srow * IN_F + colq);
        *(float4*)(ldsX + row * IN_F + colq) = xv;
        const float4 x2 = make_float4(xv.x * xv.x, xv.y * xv.y,
                                      xv.z * xv.z, xv.w * xv.w);
        *(float4*)(ldsX2 + row * IN_F + colq) = x2;
    }
    __syncthreads();

    const int lane  = tid & 31;
    const int wv    = tid >> 5;       // n-tile 0..3
    const int mrow  = lane & 15;      // A: M index / B: N index
    const int khalf = lane >> 4;      // 0 or 1 (K split across lane halves)
    const int ncol  = wv * 16 + mrow;

    v8f accQ = {};   // X@W1 tile
    v8f accP = {};   // X²@W2 tile

    for (int kk = 0; kk < IN_F; kk += 4) {
        const int kA = kk + 2 * khalf;   // lane holds K = kA, kA+1
        const v2f a1 = *(const v2f*)(ldsX  + mrow * IN_F + kA);
        const v2f a2 = *(const v2f*)(ldsX2 + mrow * IN_F + kA);
        v2f bw1, bw2;
        bw1.x = W1[(kA + 0) * IN_F + ncol];
        bw1.y = W1[(kA + 1) * IN_F + ncol];
        bw2.x = W2[(kA + 0) * IN_F + ncol];
        bw2.y = W2[(kA + 1) * IN_F + ncol];
        accQ = __builtin_amdgcn_wmma_f32_16x16x4_f32(
            false, a1, false, bw1, (short)0, accQ, false, false);
        accP = __builtin_amdgcn_wmma_f32_16x16x4_f32(
            false, a2, false, bw2, (short)0, accP, false, false);
    }

    // D layout: lane holds column ncol; VGPR v holds row rb + v + 8*khalf.
    const float bsum = b1[ncol] + b2[ncol];
#pragma unroll
    for (int v = 0; v < 8; ++v) {
        const int row = rb + v + 8 * khalf;
        if (row < nrows) {
            out[(size_t)row * IN_F + ncol] = accQ[v] + bsum;
            P  [(size_t)row * IN_F + ncol] = accQ[v] + accP[v];
        }
    }
}

// Kernel 2: out[r] += vals[e] * P[c]  (single fused SpMM pass).
// One wave per edge; each lane handles 2 features (float2 gather, 2 atomics).
__global__ __launch_bounds__(256) void gnn_spmm_scatter(
    const int* __restrict__ rows, const int* __restrict__ cols,
    const float* __restrict__ vals, const float* __restrict__ P,
    float* __restrict__ out, int n_edges)
{
    const int idx = blockIdx.x * 256 + threadIdx.x;
    const int e = idx >> 5;
    if (e >= n_edges) return;
    const int f2 = (idx & 31) * 2;
    const int   r = rows[e];
    const int   c = cols[e];
    const float v = vals[e];
    const v2f p = *(const v2f*)(P + (size_t)c * IN_F + f2);
    float* o = out + (size_t)r * IN_F + f2;
    atomicAdd(o + 0, v * p.x);
    atomicAdd(o + 1, v * p.y);
}

extern "C" void kernel_launch(void* const* d_in, const int* in_sizes, int n_in,
                              void* d_out, int out_size, void* d_ws, size_t ws_size,
                              hipStream_t stream) {
    const int*   rows = (const int*)d_in[0];
    const int*   cols = (const int*)d_in[1];
    const float* vals = (const float*)d_in[2];
    const float* X    = (const float*)d_in[3];
    const float* W1   = (const float*)d_in[4];
    const float* b1   = (const float*)d_in[5];
    const float* W2   = (const float*)d_in[6];
    const float* b2   = (const float*)d_in[7];

    const int n_edges = in_sizes[0];
    const int n_nodes = in_sizes[3] / IN_F;

    float* out = (float*)d_out;
    float* P   = (float*)d_ws;   // N * 64 f32 = 25.6 MB

    // Pass 1: dense WMMA — writes out = X@W1 + (b1+b2), P = X@W1 + X²@W2.
    const int g1 = (n_nodes + 15) / 16;
    gnn_dense_wmma<<<g1, 128, 0, stream>>>(X, W1, b1, W2, b2, out, P, n_nodes);

    // Pass 2: out += L @ P  (one wave per edge).
    const long long total = (long long)n_edges * 32;
    const int g2 = (int)((total + 255) / 256);
    gnn_spmm_scatter<<<g2, 256, 0, stream>>>(rows, cols, vals, P, out, n_edges);
}